// RadialModel_15204184228299
// MI455X (gfx1250) — compile-verified
//
#include <hip/hip_runtime.h>
#include <math.h>

// ---------------------------------------------------------------------------
// Problem constants (fixed by the reference)
// ---------------------------------------------------------------------------
#define NT      8
#define NC      8
#define NFRAME  64          // nt*nc
#define NIM     256         // image size
#define NG      512         // grid size
#define KPTS    16384
#define JW      6           // KB kernel width
#define ALPHA_F 14.04f      // 2.34 * 6
#define TWO_PI_F 6.28318530717958647692f

typedef __attribute__((ext_vector_type(2))) float v2f;
typedef __attribute__((ext_vector_type(8))) float v8f;

// workspace layout (units: floats)
static constexpr size_t NPIX_G   = (size_t)NG * NG;           // 262144
static constexpr size_t OFF_SCALE = 0;                        // 256 floats
static constexpr size_t OFF_FA_RE = 256;
static constexpr size_t OFF_FA_IM = OFF_FA_RE + NPIX_G;
static constexpr size_t OFF_FB_RE = OFF_FA_IM + NPIX_G;
static constexpr size_t OFF_FB_IM = OFF_FB_RE + NPIX_G;
static constexpr size_t OFF_S_RE  = OFF_FB_IM + NPIX_G;       // 64 frames, B-swizzled; reused as Y (row-major)
static constexpr size_t OFF_S_IM  = OFF_S_RE + NFRAME * NPIX_G;
static constexpr size_t OFF_T_RE  = OFF_S_IM + NFRAME * NPIX_G;
static constexpr size_t OFF_T_IM  = OFF_T_RE + NFRAME * NPIX_G;

// ---------------------------------------------------------------------------
// device helpers
// ---------------------------------------------------------------------------
__device__ __forceinline__ float bessel_i0(float x) {
    // power series: sum (x^2/4)^n / (n!)^2 ; converges well for x <= 14.04
    float q = 0.25f * x * x;
    float term = 1.0f, sum = 1.0f;
    for (int n = 1; n <= 32; ++n) {
        term *= q / (float)(n * n);
        sum += term;
    }
    return sum;
}

// WMMA B-operand swizzle: element (k,n) of a 512x512 B matrix -> flat index.
// B 16x4-chunk layout (assumed mirror of the documented A layout):
//   lane = n%16 + 16*((k%4)>>1), vgpr(j) = k&1, chunk kc = k/4, tile nt = n/16
__device__ __host__ __forceinline__ size_t bswz(int k, int n) {
    int kc = k >> 2, r = k & 3;
    int lane = (n & 15) | ((r >> 1) << 4);
    int ntile = n >> 4;
    return ((size_t)((kc * 32 + ntile) * 32 + lane) << 1) + (r & 1);
}

// ---------------------------------------------------------------------------
// init: apodization table  scale[n] = 1 / FT_kb((n-128)/512)
// ---------------------------------------------------------------------------
__global__ void k_init_scale(float* __restrict__ scale) {
    int n = threadIdx.x;          // 256 threads
    float f = ((float)n - 128.0f) * (1.0f / (float)NG);
    float pjf = (float)M_PI * (float)JW * f;
    float z = pjf * pjf - ALPHA_F * ALPHA_F;
    float s = sqrtf(fabsf(z));
    float val;
    if (z < 0.0f)      val = sinhf(s) / fmaxf(s, 1e-12f);
    else if (s < 1e-6f) val = 1.0f;
    else                val = sinf(s) / s;
    float ft = ((float)JW / bessel_i0(ALPHA_F)) * val;
    scale[n] = 1.0f / ft;
}

// ---------------------------------------------------------------------------
// init: DFT matrices.  FA row-major with ortho norm (1/512) folded in (A op);
// FB in WMMA-B swizzled layout, unnormalized (B op of second GEMM).
// ---------------------------------------------------------------------------
__global__ void k_init_F(float* __restrict__ fa_re, float* __restrict__ fa_im,
                         float* __restrict__ fb_re, float* __restrict__ fb_im) {
    int tid = blockIdx.x * blockDim.x + threadIdx.x;   // 262144 threads
    int r = tid >> 9, c = tid & (NG - 1);
    int ph = (r * c) & (NG - 1);
    float ang = -(TWO_PI_F / (float)NG) * (float)ph;   // e^{-2pi i rc/512}
    float cr = __cosf(ang), ci = __sinf(ang);
    fa_re[tid] = cr * (1.0f / (float)NG);
    fa_im[tid] = ci * (1.0f / (float)NG);
    size_t sw = bswz(r, c);
    fb_re[sw] = cr;
    fb_im[sw] = ci;
}

// ---------------------------------------------------------------------------
// pad: coil-combine + apodize + zero-pad + ifftshift, written directly in the
// WMMA B-swizzled layout consumed by GEMM1.  S buffers pre-zeroed by memset.
// ---------------------------------------------------------------------------
__global__ void k_pad(const float* __restrict__ x, const float* __restrict__ coil,
                      const float* __restrict__ scale,
                      float* __restrict__ s_re, float* __restrict__ s_im) {
    int tid = blockIdx.x * blockDim.x + threadIdx.x;   // 64 * 65536 threads
    int f = tid >> 16;
    int pix = tid & 65535;
    int ix = pix >> 8, iy = pix & 255;
    int t = f >> 3, c = f & 7;

    size_t ximg = ((size_t)(t * 2) * NIM + ix) * NIM + iy;
    float xr = x[ximg];
    float xi = x[ximg + (size_t)NIM * NIM];
    size_t cimg = ((size_t)(c * 2) * NIM + ix) * NIM + iy;
    float cr = coil[cimg];
    float ci = coil[cimg + (size_t)NIM * NIM];

    float s = scale[ix] * scale[iy];
    float vr = (xr * cr - xi * ci) * s;
    float vi = (xr * ci + xi * cr) * s;

    // padded position (128+ix, 128+iy) then ifftshift by 256 -> (ix+384)&511
    int u = (ix + 384) & (NG - 1);
    int v = (iy + 384) & (NG - 1);
    size_t base = (size_t)f * NPIX_G;
    size_t sw = bswz(u, v);
    s_re[base + sw] = vr;
    s_im[base + sw] = vi;
}

// ---------------------------------------------------------------------------
// Complex GEMM 512x512x512 via V_WMMA_F32_16X16X4_F32.
// One wave (32 threads) per 16x16 complex output tile; 128 K-chunks of 4.
// A: row-major planar re/im.  B: pre-swizzled planar re/im.  O: row-major.
// ---------------------------------------------------------------------------
__global__ __launch_bounds__(32)
void k_dft_gemm(const float* __restrict__ a_re, const float* __restrict__ a_im, size_t a_fstride,
                const float* __restrict__ b_re, const float* __restrict__ b_im, size_t b_fstride,
                float* __restrict__ o_re, float* __restrict__ o_im, size_t o_fstride) {
    const int lane  = threadIdx.x;          // 0..31, full EXEC (WMMA requirement)
    const int ntile = blockIdx.x;           // 0..31
    const int mtile = blockIdx.y;           // 0..31
    const size_t fz = blockIdx.z;

    const float* Ar = a_re + fz * a_fstride;
    const float* Ai = a_im + fz * a_fstride;
    const float* Br = b_re + fz * b_fstride;
    const float* Bi = b_im + fz * b_fstride;

    const int m     = lane & 15;            // A row within tile
    const int khalf = lane >> 4;            // selects K pair {0,1} vs {2,3}
    const size_t arow = (size_t)(mtile * 16 + m) * NG;

    v8f accr = {0, 0, 0, 0, 0, 0, 0, 0};
    v8f acci = {0, 0, 0, 0, 0, 0, 0, 0};

    for (int kc = 0; kc < NG / 4; ++kc) {
        const int k0 = 4 * kc + 2 * khalf;
        v2f ar = *(const v2f*)(Ar + arow + k0);
        v2f ai = *(const v2f*)(Ai + arow + k0);
        const size_t bidx = ((size_t)((kc * 32 + ntile) * 32 + lane)) << 1;
        v2f br = *(const v2f*)(Br + bidx);
        v2f bi = *(const v2f*)(Bi + bidx);
        v2f nai = -ai;                       // f32 WMMA has no A-neg modifier

        // complex: Dr += Ar*Br - Ai*Bi ; Di += Ar*Bi + Ai*Br
        accr = __builtin_amdgcn_wmma_f32_16x16x4_f32(false, ar,  false, br, (short)0, accr, false, false);
        accr = __builtin_amdgcn_wmma_f32_16x16x4_f32(false, nai, false, bi, (short)0, accr, false, false);
        acci = __builtin_amdgcn_wmma_f32_16x16x4_f32(false, ar,  false, bi, (short)0, acci, false, false);
        acci = __builtin_amdgcn_wmma_f32_16x16x4_f32(false, ai,  false, br, (short)0, acci, false, false);
    }

    // D layout: vgpr v -> M = v (lanes 0-15) / v+8 (lanes 16-31); N = lane&15
    float* Or = o_re + fz * o_fstride;
    float* Oi = o_im + fz * o_fstride;
    const int ng = ntile * 16 + (lane & 15);
#pragma unroll
    for (int v = 0; v < 8; ++v) {
        int mg = mtile * 16 + v + 8 * khalf;
        Or[(size_t)mg * NG + ng] = accr[v];
        Oi[(size_t)mg * NG + ng] = acci[v];
    }
}

// ---------------------------------------------------------------------------
// KB gridding interpolation + sqrt(w).  One thread per (t, kpoint); weights
// computed once and reused across the 8 coils (grids are L2-resident).
// ---------------------------------------------------------------------------
__global__ void k_interp(const float* __restrict__ ktraj, const float* __restrict__ w,
                         const float* __restrict__ y_re, const float* __restrict__ y_im,
                         float* __restrict__ out) {
    int tid = blockIdx.x * blockDim.x + threadIdx.x;   // NT*KPTS threads
    int kk = tid & (KPTS - 1);
    int t  = tid >> 14;

    float omx = ktraj[((size_t)(t * 2) + 0) * KPTS + kk];
    float omy = ktraj[((size_t)(t * 2) + 1) * KPTS + kk];

    float gx = fmodf(omx, TWO_PI_F); if (gx < 0.0f) gx += TWO_PI_F;
    float gy = fmodf(omy, TWO_PI_F); if (gy < 0.0f) gy += TWO_PI_F;
    gx *= (float)NG / TWO_PI_F;
    gy *= (float)NG / TWO_PI_F;

    const float i0a_inv = 1.0f / bessel_i0(ALPHA_F);
    float wx[JW], wy[JW];
    int   ixs[JW], iys[JW];

    float mx0 = floorf(gx - 0.5f * (float)JW);
    float my0 = floorf(gy - 0.5f * (float)JW);
#pragma unroll
    for (int a = 0; a < JW; ++a) {
        float mm = mx0 + (float)(a + 1);
        float u = (gx - mm) * (2.0f / (float)JW);     // 2u/J
        float tt = 1.0f - u * u;
        wx[a] = (tt > 0.0f) ? bessel_i0(ALPHA_F * sqrtf(tt)) * i0a_inv : 0.0f;
        ixs[a] = ((int)mm + NG) & (NG - 1);
        mm = my0 + (float)(a + 1);
        u = (gy - mm) * (2.0f / (float)JW);
        tt = 1.0f - u * u;
        wy[a] = (tt > 0.0f) ? bessel_i0(ALPHA_F * sqrtf(tt)) * i0a_inv : 0.0f;
        iys[a] = ((int)mm + NG) & (NG - 1);
    }

    for (int c = 0; c < NC; ++c) {
        const size_t fbase = (size_t)(t * NC + c) * NPIX_G;
        const float* yr = y_re + fbase;
        const float* yi = y_im + fbase;
        float re = 0.0f, im = 0.0f;
#pragma unroll
        for (int a = 0; a < JW; ++a) {
            const int rowoff = ixs[a] * NG;
            const float wxa = wx[a];
#pragma unroll
            for (int b = 0; b < JW; ++b) {
                const float ww = wxa * wy[b];
                const int idx = rowoff + iys[b];
                re += yr[idx] * ww;
                im += yi[idx] * ww;
            }
        }
        size_t obase = ((size_t)(t * NC + c) * 2) * KPTS + kk;
        out[obase]        = re * sqrtf(w[obase]);
        out[obase + KPTS] = im * sqrtf(w[obase + KPTS]);
    }
}

// ---------------------------------------------------------------------------
// host launcher
// ---------------------------------------------------------------------------
extern "C" void kernel_launch(void* const* d_in, const int* in_sizes, int n_in,
                              void* d_out, int out_size, void* d_ws, size_t ws_size,
                              hipStream_t stream) {
    (void)in_sizes; (void)n_in; (void)out_size; (void)ws_size;

    const float* x    = (const float*)d_in[0];   // (8,2,256,256)
    const float* kt   = (const float*)d_in[1];   // (8,2,16384)
    const float* coil = (const float*)d_in[2];   // (1,8,2,256,256)
    const float* w    = (const float*)d_in[3];   // (8,8,2,16384)
    float* out = (float*)d_out;                  // (8,8,2,16384)
    float* W   = (float*)d_ws;

    float* scale = W + OFF_SCALE;
    float* fa_re = W + OFF_FA_RE;  float* fa_im = W + OFF_FA_IM;
    float* fb_re = W + OFF_FB_RE;  float* fb_im = W + OFF_FB_IM;
    float* s_re  = W + OFF_S_RE;   float* s_im  = W + OFF_S_IM;   // also Y after GEMM2
    float* t_re  = W + OFF_T_RE;   float* t_im  = W + OFF_T_IM;

    // zero the padded (B-swizzled) input grids: S_re and S_im are contiguous
    hipMemsetAsync(s_re, 0, (size_t)2 * NFRAME * NPIX_G * sizeof(float), stream);

    k_init_scale<<<1, 256, 0, stream>>>(scale);
    k_init_F<<<NPIX_G / 256, 256, 0, stream>>>(fa_re, fa_im, fb_re, fb_im);

    k_pad<<<(NFRAME * 65536) / 256, 256, 0, stream>>>(x, coil, scale, s_re, s_im);

    // T = F * S   (A = FA with ortho norm, frame-invariant; B = per-frame S)
    k_dft_gemm<<<dim3(NG / 16, NG / 16, NFRAME), 32, 0, stream>>>(
        fa_re, fa_im, (size_t)0, s_re, s_im, NPIX_G, t_re, t_im, NPIX_G);
    // Y = T * F   (A = per-frame T; B = FB, frame-invariant); Y reuses S region
    k_dft_gemm<<<dim3(NG / 16, NG / 16, NFRAME), 32, 0, stream>>>(
        t_re, t_im, NPIX_G, fb_re, fb_im, (size_t)0, s_re, s_im, NPIX_G);

    k_interp<<<(NT * KPTS) / 256, 256, 0, stream>>>(kt, w, s_re, s_im, out);
}